// CAGU_19198503813544
// MI455X (gfx1250) — compile-verified
//
#include <hip/hip_runtime.h>
#include <hip/hip_bf16.h>

#define NB   16      // batch
#define NN   512     // nodes
#define NF   625     // features (25*25)
#define NFH  937     // hidden features
#define FP   640     // NF padded  (mult of 32, = 5*128)
#define FHP  960     // NFH padded (mult of 32)
#define RR   (NB * NN)   // 8192 flattened rows
#define TI   8       // i-rows per adjacency block

typedef float v2f __attribute__((ext_vector_type(2)));
typedef float v8f __attribute__((ext_vector_type(8)));

// ---------------------------------------------------------------------------
// Zero-padding copy: dst[dstR, dstC] <- src[srcR, srcC], pad with 0.
// Used once per launch for X / weights / biases so the GEMM hot loops carry
// no bounds checks (pads multiply by zero and never perturb results).
// ---------------------------------------------------------------------------
__global__ __launch_bounds__(256) void pad2d_kernel(
    const float* __restrict__ src, float* __restrict__ dst,
    int srcR, int srcC, int dstR, int dstC)
{
    int idx = blockIdx.x * 256 + threadIdx.x;
    if (idx >= dstR * dstC) return;
    int r = idx / dstC, c = idx % dstC;
    dst[idx] = (r < srcR && c < srcC) ? src[r * srcC + c] : 0.0f;
}

// ---------------------------------------------------------------------------
// Adjacency + D^-1(A+I), one pass. Block = (b, 8 consecutive i rows).
// The 8 xi rows are one contiguous 20KB span -> coalesced LDS fill. Each
// thread owns 2 j-columns and streams each xj row ONCE while accumulating all
// 16 (i,j) L1 distances => 8x less L2 traffic than one-i-per-block.
// ---------------------------------------------------------------------------
__global__ __launch_bounds__(256) void adj_norm_kernel(
    const float* __restrict__ X,    // [B, N, NF]
    const float* __restrict__ att,  // [B, N]
    float* __restrict__ Lout)       // [B, N, N]
{
    const int b  = blockIdx.y;
    const int i0 = blockIdx.x * TI;
    __shared__ float xi[TI * NF];   // 20000 B
    __shared__ float red[256];
    __shared__ float sdinv[TI];

    const float* Xb = X + (size_t)b * NN * NF;
    for (int idx = threadIdx.x; idx < TI * NF; idx += 256)
        xi[idx] = Xb[(size_t)i0 * NF + idx];   // rows i0..i0+7 are contiguous
    __syncthreads();

    const int j0 = threadIdx.x;
    const int j1 = threadIdx.x + 256;
    const float* xj0 = Xb + (size_t)j0 * NF;
    const float* xj1 = Xb + (size_t)j1 * NF;

    float d[TI][2];
    #pragma unroll
    for (int r = 0; r < TI; ++r) { d[r][0] = 0.0f; d[r][1] = 0.0f; }

    #pragma unroll 2
    for (int f = 0; f < NF; ++f) {
        const float v0 = xj0[f];
        const float v1 = xj1[f];
        #pragma unroll
        for (int r = 0; r < TI; ++r) {
            const float xf = xi[r * NF + f];       // uniform LDS broadcast
            d[r][0] += fabsf(xf - v0);
            d[r][1] += fabsf(xf - v1);
        }
    }

    const float aj0 = att[b * NN + j0];
    const float aj1 = att[b * NN + j1];

    float aval[TI][2];
    #pragma unroll
    for (int r = 0; r < TI; ++r) {
        const float air = att[b * NN + i0 + r];
        float a0 = ((d[r][0] <= 180.0f) && (fabsf(air - aj0) <= 0.05f)) ? 1.0f : 0.0f;
        float a1 = ((d[r][1] <= 180.0f) && (fabsf(air - aj1) <= 0.05f)) ? 1.0f : 0.0f;
        if (j0 == i0 + r) a0 = 1.0f;   // diag zeroed, then +I
        if (j1 == i0 + r) a1 = 1.0f;
        aval[r][0] = a0;
        aval[r][1] = a1;
    }

    // per-row degree -> 1/deg
    for (int r = 0; r < TI; ++r) {
        red[threadIdx.x] = aval[r][0] + aval[r][1];
        __syncthreads();
        for (int s = 128; s > 0; s >>= 1) {
            if (threadIdx.x < s) red[threadIdx.x] += red[threadIdx.x + s];
            __syncthreads();
        }
        if (threadIdx.x == 0) {
            const float dg = red[0];
            sdinv[r] = (dg > 0.0f) ? (1.0f / dg) : 0.0f;
        }
        __syncthreads();
    }

    #pragma unroll
    for (int r = 0; r < TI; ++r) {
        float* Lrow = Lout + ((size_t)b * NN + (i0 + r)) * NN;
        Lrow[j0] = aval[r][0] * sdinv[r];
        Lrow[j1] = aval[r][1] * sdinv[r];
    }
}

// ---------------------------------------------------------------------------
// WMMA f32 16x16x4 helpers. Layout (CDNA5 ISA 7.12.2):
//   A 16x4: lanes 0-15 hold K={k0,k0+1}, lanes 16-31 hold K={k0+2,k0+3}
//   C/D:    VGPR r = row r (lanes 0-15) / row r+8 (lanes 16-31)
// No bounds checks: operands are padded, k strides 8B-aligned -> b64 merges.
// ---------------------------------------------------------------------------
__device__ __forceinline__ v8f wmma4(v2f a, v2f b, v8f c) {
    return __builtin_amdgcn_wmma_f32_16x16x4_f32(
        false, a, false, b, (short)0, c, false, false);
}

// ---------------------------------------------------------------------------
// Batched dense  T[b] = L[b] @ M[b].   L: [NN,NN], M: [NN,Cp], K = NN = 512.
// 8 waves/block as 2(row)x4(col); 32x32 tile per wave (4 accumulators,
// 2 A-frags x 2 B-frags per k-step -> 8 flops/byte from L2).
// ---------------------------------------------------------------------------
__global__ __launch_bounds__(256) void spmm_kernel(
    const float* __restrict__ Lm,   // [B, NN, NN]
    const float* __restrict__ Mm,   // [B, NN, Cp] (padded cols are zero)
    float* __restrict__ Out,        // [B, NN, Cp]
    int Cp)
{
    const int lane = threadIdx.x & 31;
    const int wave = threadIdx.x >> 5;
    const int b    = blockIdx.z;
    const int row0 = blockIdx.y * 64 + (wave & 1) * 32;
    const int col0 = blockIdx.x * 128 + (wave >> 1) * 32;
    if (col0 >= Cp) return;                      // wave-uniform: EXEC stays all-1

    const int mlo   = lane & 15;
    const int khalf = (lane >> 4) * 2;

    const float* A = Lm + (size_t)b * NN * NN;
    const float* B = Mm + (size_t)b * NN * Cp;

    v8f acc00 = {}, acc01 = {}, acc10 = {}, acc11 = {};

    #pragma unroll 2
    for (int k0 = 0; k0 < NN; k0 += 4) {
        const int kk = k0 + khalf;
        const float* ap = A + (size_t)(row0 + mlo) * NN + kk;
        v2f a0 = { ap[0],        ap[1]        };          // global_load_b64
        v2f a1 = { ap[16 * NN],  ap[16 * NN + 1] };
        const float* bp = B + (size_t)kk * Cp + col0 + mlo;
        v2f b0 = { bp[0],  bp[Cp]      };
        v2f b1 = { bp[16], bp[Cp + 16] };
        acc00 = wmma4(a0, b0, acc00);
        acc01 = wmma4(a0, b1, acc01);
        acc10 = wmma4(a1, b0, acc10);
        acc11 = wmma4(a1, b1, acc11);
    }

    float* o = Out + (size_t)b * NN * Cp;
    const int mbase = row0 + 8 * (lane >> 4);
    #pragma unroll
    for (int r = 0; r < 8; ++r) {
        const int m = mbase + r;
        o[(size_t)m        * Cp + col0 + mlo]      = acc00[r];
        o[(size_t)m        * Cp + col0 + 16 + mlo] = acc01[r];
        o[(size_t)(m + 16) * Cp + col0 + mlo]      = acc10[r];
        o[(size_t)(m + 16) * Cp + col0 + 16 + mlo] = acc11[r];
    }
}

// ---------------------------------------------------------------------------
// Fused ChebConv:  Out = relu(Xa @ W0 + Ta @ W1 + bias)
// Xa, Ta: [RR, Kp] (lda == Kp, padded); W: [Kp, Cp] (zero-padded).
// Store masked only in the epilogue (final layer writes exact [RR, NF]).
// ---------------------------------------------------------------------------
__global__ __launch_bounds__(256) void cheb_kernel(
    const float* __restrict__ Xa,
    const float* __restrict__ Ta,
    const float* __restrict__ W0,
    const float* __restrict__ W1p,
    const float* __restrict__ bias, // [Cp] zero-padded
    float* __restrict__ Out,        // [RR, ldo]
    int Kp, int Cp, int Cout, int ldo)
{
    const int lane = threadIdx.x & 31;
    const int wave = threadIdx.x >> 5;
    const int row0 = blockIdx.y * 64 + (wave & 1) * 32;
    const int col0 = blockIdx.x * 128 + (wave >> 1) * 32;
    if (col0 >= Cp) return;                      // wave-uniform

    const int mlo   = lane & 15;
    const int khalf = (lane >> 4) * 2;

    v8f acc00 = {}, acc01 = {}, acc10 = {}, acc11 = {};

    #pragma unroll 2
    for (int k0 = 0; k0 < Kp; k0 += 4) {
        const int kk = k0 + khalf;
        {   // T0 term: Xa @ W0
            const float* ap = Xa + (size_t)(row0 + mlo) * Kp + kk;
            v2f a0 = { ap[0],       ap[1]       };
            v2f a1 = { ap[16 * Kp], ap[16 * Kp + 1] };
            const float* bp = W0 + (size_t)kk * Cp + col0 + mlo;
            v2f b0 = { bp[0],  bp[Cp]      };
            v2f b1 = { bp[16], bp[Cp + 16] };
            acc00 = wmma4(a0, b0, acc00);
            acc01 = wmma4(a0, b1, acc01);
            acc10 = wmma4(a1, b0, acc10);
            acc11 = wmma4(a1, b1, acc11);
        }
        {   // T1 term: Ta @ W1
            const float* ap = Ta + (size_t)(row0 + mlo) * Kp + kk;
            v2f a0 = { ap[0],       ap[1]       };
            v2f a1 = { ap[16 * Kp], ap[16 * Kp + 1] };
            const float* bp = W1p + (size_t)kk * Cp + col0 + mlo;
            v2f b0 = { bp[0],  bp[Cp]      };
            v2f b1 = { bp[16], bp[Cp + 16] };
            acc00 = wmma4(a0, b0, acc00);
            acc01 = wmma4(a0, b1, acc01);
            acc10 = wmma4(a1, b0, acc10);
            acc11 = wmma4(a1, b1, acc11);
        }
    }

    const int n0 = col0 + mlo;
    const int n1 = col0 + 16 + mlo;
    const float bv0 = bias[n0];
    const float bv1 = bias[n1];
    const bool ok0 = (n0 < Cout);
    const bool ok1 = (n1 < Cout);
    const int mbase = row0 + 8 * (lane >> 4);

    #pragma unroll
    for (int r = 0; r < 8; ++r) {
        const int m = mbase + r;
        if (ok0) {
            Out[(size_t)m        * ldo + n0] = fmaxf(acc00[r] + bv0, 0.0f);
            Out[(size_t)(m + 16) * ldo + n0] = fmaxf(acc10[r] + bv0, 0.0f);
        }
        if (ok1) {
            Out[(size_t)m        * ldo + n1] = fmaxf(acc01[r] + bv1, 0.0f);
            Out[(size_t)(m + 16) * ldo + n1] = fmaxf(acc11[r] + bv1, 0.0f);
        }
    }
}

// ---------------------------------------------------------------------------
static inline int cdiv(int a, int b) { return (a + b - 1) / b; }

extern "C" void kernel_launch(void* const* d_in, const int* in_sizes, int n_in,
                              void* d_out, int out_size, void* d_ws, size_t ws_size,
                              hipStream_t stream) {
    const float* x4  = (const float*)d_in[0];  // [8192, 625]
    const float* att = (const float*)d_in[1];  // [16, 512]
    const float* W1  = (const float*)d_in[2];  // [2, 625, 937]
    const float* b1  = (const float*)d_in[3];  // [937]
    const float* W2  = (const float*)d_in[4];  // [2, 937, 625]
    const float* b2  = (const float*)d_in[5];  // [625]
    float* out = (float*)d_out;                // [8192, 625]

    // Workspace (floats), total ~132 MB -> fully L2-resident on MI455X (192MB)
    float* p    = (float*)d_ws;
    float* Xp   = p;  p += (size_t)RR * FP;    // [8192, 640]
    float* W1p0 = p;  p += (size_t)FP * FHP;   // [640, 960]
    float* W1p1 = p;  p += (size_t)FP * FHP;
    float* b1p  = p;  p += FHP;
    float* W2p0 = p;  p += (size_t)FHP * FP;   // [960, 640]
    float* W2p1 = p;  p += (size_t)FHP * FP;
    float* b2p  = p;  p += FP;
    float* L    = p;  p += (size_t)NB * NN * NN;  // [16,512,512]
    float* T1   = p;  p += (size_t)RR * FP;       // [8192, 640]
    float* h    = p;  p += (size_t)RR * FHP;      // [8192, 960]
    float* T2   = p;  /* [8192, 960] */

    // --- pack / zero-pad operands (one-time, bandwidth-trivial) ---
    pad2d_kernel<<<cdiv(RR * FP, 256), 256, 0, stream>>>(x4, Xp, RR, NF, RR, FP);
    pad2d_kernel<<<cdiv(FP * FHP, 256), 256, 0, stream>>>(W1, W1p0, NF, NFH, FP, FHP);
    pad2d_kernel<<<cdiv(FP * FHP, 256), 256, 0, stream>>>(W1 + (size_t)NF * NFH, W1p1, NF, NFH, FP, FHP);
    pad2d_kernel<<<cdiv(FHP, 256), 256, 0, stream>>>(b1, b1p, 1, NFH, 1, FHP);
    pad2d_kernel<<<cdiv(FHP * FP, 256), 256, 0, stream>>>(W2, W2p0, NFH, NF, FHP, FP);
    pad2d_kernel<<<cdiv(FHP * FP, 256), 256, 0, stream>>>(W2 + (size_t)NFH * NF, W2p1, NFH, NF, FHP, FP);
    pad2d_kernel<<<cdiv(FP, 256), 256, 0, stream>>>(b2, b2p, 1, NF, 1, FP);

    // --- 1) L = D^-1 (A + I) ---
    adj_norm_kernel<<<dim3(NN / TI, NB), 256, 0, stream>>>(x4, att, L);

    // --- 2) T1 = L @ Xp  (Cp = 640) ---
    spmm_kernel<<<dim3(FP / 128, NN / 64, NB), 256, 0, stream>>>(L, Xp, T1, FP);

    // --- 3) h = relu(Xp @ W1p0 + T1 @ W1p1 + b1p)  (Kp=640, Cp=960) ---
    cheb_kernel<<<dim3(cdiv(FHP, 128), RR / 64), 256, 0, stream>>>(
        Xp, T1, W1p0, W1p1, b1p, h, FP, FHP, FHP, FHP);

    // --- 4) T2 = L @ h  (Cp = 960) ---
    spmm_kernel<<<dim3(cdiv(FHP, 128), NN / 64, NB), 256, 0, stream>>>(L, h, T2, FHP);

    // --- 5) out = relu(h @ W2p0 + T2 @ W2p1 + b2p), exact [8192,625] store ---
    cheb_kernel<<<dim3(FP / 128, RR / 64), 256, 0, stream>>>(
        h, T2, W2p0, W2p1, b2p, out, FHP, FP, NF, NF);
}